// Stem_19731079758368
// MI455X (gfx1250) — compile-verified
//
#include <hip/hip_runtime.h>
#include <hip/hip_bf16.h>
#include <math.h>

// ---------------------------------------------------------------------------
// CDNA5 (gfx1250) fp32 WMMA: V_WMMA_F32_16X16X4_F32
//   A: 16x4 f32  -> 2 VGPRs/lane. lanes 0-15: M=lane, K={k,k+1};
//                                 lanes 16-31: M=lane-16, K={k+2,k+3}
//   B: 4x16 f32  -> symmetric: lanes 0-15: N=lane, K={k,k+1};
//                              lanes 16-31: N=lane-16, K={k+2,k+3}
//   C/D: 16x16 f32 -> 8 VGPRs: element (M,N): lane = N + 16*(M>=8), vgpr = M%8
// ---------------------------------------------------------------------------
typedef float v2f __attribute__((ext_vector_type(2)));
typedef float v8f __attribute__((ext_vector_type(8)));

__device__ __forceinline__ v8f wmma_f32_k4(v2f a, v2f b, v8f c) {
  // 8 args: (neg_a, A, neg_b, B, c_mod, C, reuse_a, reuse_b)
  return __builtin_amdgcn_wmma_f32_16x16x4_f32(false, a, false, b, (short)0, c,
                                               false, false);
}

__device__ __forceinline__ float sigm(float x) { return 1.0f / (1.0f + __expf(-x)); }

#define BB   128          // batch
#define DD   512          // input dim
#define HH   256          // hidden
#define G4H  1024         // 4H
#define LMAX 64
#define LTOT 2080         // sum_{l=1..64} l
// ---------------------------------------------------------------------------
// uh0 = unit_hidden * existence; ex0 = existence as float
// ---------------------------------------------------------------------------
__global__ void k_init(const unsigned char* __restrict__ exist,
                       const float* __restrict__ uhin,
                       float* __restrict__ uh0, float* __restrict__ ex0) {
  size_t idx = (size_t)blockIdx.x * blockDim.x + threadIdx.x;
  const size_t total = (size_t)BB * LMAX * DD;
  if (idx >= total) return;
  int d   = (int)(idx % DD);
  int pos = (int)((idx / DD) % LMAX);
  int b   = (int)(idx / ((size_t)DD * LMAX));
  unsigned char e = exist[b * LMAX + pos];
  uh0[idx] = e ? uhin[idx] : 0.0f;
  if (d == 0) ex0[b * LMAX + pos] = e ? 1.0f : 0.0f;
}

// ---------------------------------------------------------------------------
// Input projection GEMM: C[M,2048] = A[M,512] x W^T + bias
//   cols [0,1024)  : w_ih_f / b_f (forward gates i,f,g,o)
//   cols [1024,2048): w_ih_b / b_b (backward gates)
// One wave (32 thr) computes a 16(M) x 64(N) tile: 4 accumulators share A frag.
// ---------------------------------------------------------------------------
__global__ __launch_bounds__(32) void k_gemm_xg(
    const float* __restrict__ A,    // [M, 512]
    const float* __restrict__ Wf,   // [1024, 512]
    const float* __restrict__ Wb,   // [1024, 512]
    const float* __restrict__ bf,   // [1024]
    const float* __restrict__ bb,   // [1024]
    float* __restrict__ C,          // [M, 2048]
    int M) {
  const int lane = threadIdx.x;
  const int hi = lane >> 4;         // which K-pair this half-wave holds
  const int lo = lane & 15;
  const int m0 = blockIdx.x * 16;
  const int n0 = blockIdx.y * 64;   // 0..2047, step 64 (never straddles 1024)
  const float* W;  const float* bias;  int nw0;
  if (n0 < G4H) { W = Wf; bias = bf; nw0 = n0; }
  else          { W = Wb; bias = bb; nw0 = n0 - G4H; }

  v8f acc[4];
#pragma unroll
  for (int s = 0; s < 4; ++s) {
    float bv = bias[nw0 + s * 16 + lo];
    acc[s] = (v8f){bv, bv, bv, bv, bv, bv, bv, bv};
  }

  const float* arow = A + (size_t)(m0 + lo) * DD;
  for (int k = 0; k < DD; k += 4) {
    const int ka = k + 2 * hi;
    if (k + 64 < DD) __builtin_prefetch(arow + ka + 64, 0, 1);  // global_prefetch
    v2f a;  a.x = arow[ka];  a.y = arow[ka + 1];
#pragma unroll
    for (int s = 0; s < 4; ++s) {
      const float* wrow = W + (size_t)(nw0 + s * 16 + lo) * DD;
      v2f bv;  bv.x = wrow[ka];  bv.y = wrow[ka + 1];
      acc[s] = wmma_f32_k4(a, bv, acc[s]);
    }
  }

#pragma unroll
  for (int s = 0; s < 4; ++s) {
    const int n = n0 + s * 16 + lo;
#pragma unroll
    for (int j = 0; j < 8; ++j) {
      const int m = m0 + j + 8 * hi;
      C[(size_t)m * 2048 + n] = acc[s][j];
    }
  }
}

// ---------------------------------------------------------------------------
// One LSTM time step, both directions fused (blockIdx.z = dir).
// Wave computes 16(batch) x 16(hidden) for ALL 4 gates (4 WMMA accumulators
// seeded from xg), then does the LSTM pointwise in registers.
// h-state ping-pongs (cross-block RAW across launches); c-state is block-local.
// ---------------------------------------------------------------------------
__global__ __launch_bounds__(32) void k_lstm_step(
    const float* __restrict__ xg,     // [B*l, 2048]
    const float* __restrict__ whh_f,  // [1024, 256]
    const float* __restrict__ whh_b,  // [1024, 256]
    const float* __restrict__ h0,     // [2, 256] broadcast initial h
    const float* __restrict__ c0,     // [2, 256] broadcast initial c
    const float* __restrict__ h_in,   // [2, B, 256] (unused at t==0)
    float* __restrict__ h_out,        // [2, B, 256]
    float* __restrict__ c_state,      // [2, B, 256]
    float* __restrict__ hs,           // [B, l, 512]
    int l, int t) {
  const int lane = threadIdx.x;
  const int hi = lane >> 4;
  const int lo = lane & 15;
  const int m0  = blockIdx.x * 16;     // batch tile
  const int hb  = blockIdx.y * 16;     // hidden-col tile
  const int dir = blockIdx.z;
  const int pos = dir ? (l - 1 - t) : t;
  const float* W = dir ? whh_b : whh_f;

  // seed accumulators with xg (input-gate contributions incl. bias)
  v8f acc[4];
#pragma unroll
  for (int q = 0; q < 4; ++q) {
    const int n = dir * G4H + q * HH + hb + lo;
#pragma unroll
    for (int j = 0; j < 8; ++j) {
      const int b = m0 + j + 8 * hi;
      acc[q][j] = xg[(size_t)(b * l + pos) * 2048 + n];
    }
  }

  // += h_prev @ w_hh^T   (K = 256)
  const float* hprev = h_in + (size_t)dir * BB * HH;
  for (int k = 0; k < HH; k += 4) {
    const int ka = k + 2 * hi;
    v2f a;
    if (t == 0) { a.x = h0[dir * HH + ka];  a.y = h0[dir * HH + ka + 1]; }
    else {
      const float* ar = hprev + (size_t)(m0 + lo) * HH;
      a.x = ar[ka];  a.y = ar[ka + 1];
    }
#pragma unroll
    for (int q = 0; q < 4; ++q) {
      const float* wr = W + (size_t)(q * HH + hb + lo) * HH;
      v2f bv;  bv.x = wr[ka];  bv.y = wr[ka + 1];
      acc[q] = wmma_f32_k4(a, bv, acc[q]);
    }
  }

  // LSTM pointwise (torch gate order i,f,g,o), in registers
  const int hcol = hb + lo;
#pragma unroll
  for (int j = 0; j < 8; ++j) {
    const int b = m0 + j + 8 * hi;
    const size_t sidx = ((size_t)dir * BB + b) * HH + hcol;
    float cold = (t == 0) ? c0[dir * HH + hcol] : c_state[sidx];
    float ig = sigm(acc[0][j]);
    float fg = sigm(acc[1][j]);
    float gg = tanhf(acc[2][j]);
    float og = sigm(acc[3][j]);
    float cn = fg * cold + ig * gg;
    float hn = og * tanhf(cn);
    c_state[sidx] = cn;
    h_out[sidx]   = hn;
    hs[(size_t)(b * l + pos) * 512 + dir * HH + hcol] = hn;
  }
}

// ---------------------------------------------------------------------------
// Orientation: o[b,pos] = dot(hs[b,pos,:512], orient_w) + orient_b
// One wave per (b,pos); wave32 shuffle reduction. Also emits output slices.
// ---------------------------------------------------------------------------
__global__ __launch_bounds__(256) void k_orient(
    const float* __restrict__ hs,     // [B, l, 512]
    const float* __restrict__ ow,     // [512]
    const float* __restrict__ ob,     // [1]
    const float* __restrict__ ex,     // [B, l] 0/1
    float* __restrict__ o_buf,        // [B, l]
    float* __restrict__ orients_out,  // [B, 2080]
    float* __restrict__ exs_out,      // [B, 2080]
    int l, int out_off) {
  const int wave = (int)((blockIdx.x * blockDim.x + threadIdx.x) >> 5);
  const int lane = threadIdx.x & 31;
  const int total = BB * l;
  if (wave >= total) return;            // uniform per wave32
  const int b = wave / l, pos = wave % l;
  const float* row = hs + (size_t)(b * l + pos) * 512;
  float s = 0.0f;
  for (int k = lane; k < 512; k += 32) s += row[k] * ow[k];
#pragma unroll
  for (int off = 16; off > 0; off >>= 1) s += __shfl_xor(s, off, 32);
  if (lane == 0) {
    const float o = s + ob[0];
    o_buf[b * l + pos] = o;
    orients_out[(size_t)b * LTOT + out_off + pos] = o;
    exs_out[(size_t)b * LTOT + out_off + pos] = ex[b * l + pos];
  }
}

// ---------------------------------------------------------------------------
// Emit units slice; merge adjacent units into the next (shorter) sequence.
//   lw = (o[p] > 0) & ex[p] ;  rw = !(o[p+1] > 0) & ex[p+1]
// ---------------------------------------------------------------------------
__global__ __launch_bounds__(256) void k_merge(
    const float* __restrict__ uh,      // [B, l, 512]
    const float* __restrict__ ex,      // [B, l]
    const float* __restrict__ o_buf,   // [B, l]
    float* __restrict__ units_out,     // [B, 2080, 512]
    float* __restrict__ uh_next,       // [B, l-1, 512]
    float* __restrict__ ex_next,       // [B, l-1]
    int l, int out_off) {
  size_t idx = (size_t)blockIdx.x * blockDim.x + threadIdx.x;
  const size_t total = (size_t)BB * l * DD;
  if (idx >= total) return;
  const int d   = (int)(idx % DD);
  const int pos = (int)((idx / DD) % l);
  const int b   = (int)(idx / ((size_t)DD * l));
  const float u = uh[idx];
  units_out[((size_t)b * LTOT + out_off + pos) * DD + d] = u;
  if (pos < l - 1) {
    const float oL = o_buf[b * l + pos];
    const float oR = o_buf[b * l + pos + 1];
    const float lw = (oL >  0.0f && ex[b * l + pos]     != 0.0f) ? 1.0f : 0.0f;
    const float rw = (oR <= 0.0f && ex[b * l + pos + 1] != 0.0f) ? 1.0f : 0.0f;
    const float uR = uh[((size_t)(b * l + pos + 1)) * DD + d];
    uh_next[((size_t)(b * (l - 1) + pos)) * DD + d] = u * lw + uR * rw;
    if (d == 0) ex_next[b * (l - 1) + pos] = (lw + rw > 0.0f) ? 1.0f : 0.0f;
  }
}

// ---------------------------------------------------------------------------
extern "C" void kernel_launch(void* const* d_in, const int* in_sizes, int n_in,
                              void* d_out, int out_size, void* d_ws, size_t ws_size,
                              hipStream_t stream) {
  (void)in_sizes; (void)n_in; (void)out_size; (void)ws_size;
  const unsigned char* existence = (const unsigned char*)d_in[0];
  const float* unit_hidden = (const float*)d_in[1];
  const float* w_ih_f = (const float*)d_in[2];
  const float* w_hh_f = (const float*)d_in[3];
  const float* b_f    = (const float*)d_in[4];
  const float* w_ih_b = (const float*)d_in[5];
  const float* w_hh_b = (const float*)d_in[6];
  const float* b_b    = (const float*)d_in[7];
  const float* ow     = (const float*)d_in[8];   // [1,512]
  const float* ob     = (const float*)d_in[9];   // [1]
  const float* h0     = (const float*)d_in[10];  // [2,1,256]
  const float* c0     = (const float*)d_in[11];  // [2,1,256]

  float* out = (float*)d_out;
  float* units_out   = out;                                    // [B,2080,512]
  float* orients_out = out + (size_t)BB * LTOT * DD;           // [B,2080]
  float* exs_out     = orients_out + (size_t)BB * LTOT;        // [B,2080]

  // workspace carve-up
  float* p = (float*)d_ws;
  float* uhA = p;  p += (size_t)BB * LMAX * DD;
  float* uhB = p;  p += (size_t)BB * LMAX * DD;
  float* xg  = p;  p += (size_t)BB * LMAX * 2048;
  float* hs  = p;  p += (size_t)BB * LMAX * 512;
  float* hS0 = p;  p += (size_t)2 * BB * HH;
  float* hS1 = p;  p += (size_t)2 * BB * HH;
  float* cS  = p;  p += (size_t)2 * BB * HH;
  float* exA = p;  p += (size_t)BB * LMAX;
  float* exB = p;  p += (size_t)BB * LMAX;
  float* obuf = p; p += (size_t)BB * LMAX;

  {
    size_t total = (size_t)BB * LMAX * DD;
    k_init<<<(unsigned)((total + 255) / 256), 256, 0, stream>>>(
        existence, unit_hidden, uhA, exA);
  }

  float* uh_cur = uhA; float* uh_nxt = uhB;
  float* ex_cur = exA; float* ex_nxt = exB;
  float* hbuf[2] = {hS0, hS1};

  for (int l = LMAX; l >= 1; --l) {
    const int M = BB * l;
    const int out_off = l * (l - 1) / 2;   // ascending-length concat offset

    dim3 g1(M / 16, 2048 / 64);
    k_gemm_xg<<<g1, 32, 0, stream>>>(uh_cur, w_ih_f, w_ih_b, b_f, b_b, xg, M);

    for (int t = 0; t < l; ++t) {
      k_lstm_step<<<dim3(BB / 16, HH / 16, 2), 32, 0, stream>>>(
          xg, w_hh_f, w_hh_b, h0, c0,
          hbuf[t & 1], hbuf[(t + 1) & 1], cS, hs, l, t);
    }

    {
      int waves = BB * l;
      size_t threads = (size_t)waves * 32;
      k_orient<<<(unsigned)((threads + 255) / 256), 256, 0, stream>>>(
          hs, ow, ob, ex_cur, obuf, orients_out, exs_out, l, out_off);
    }
    {
      size_t total = (size_t)BB * l * DD;
      k_merge<<<(unsigned)((total + 255) / 256), 256, 0, stream>>>(
          uh_cur, ex_cur, obuf, units_out, uh_nxt, ex_nxt, l, out_off);
    }

    float* tu = uh_cur; uh_cur = uh_nxt; uh_nxt = tu;
    float* te = ex_cur; ex_cur = ex_nxt; ex_nxt = te;
  }
}